// ScaledDotProductAttention_1640677507314
// MI455X (gfx1250) — compile-verified
//
#include <hip/hip_runtime.h>
#include <hip/hip_bf16.h>

typedef __attribute__((ext_vector_type(2))) float v2f;
typedef __attribute__((ext_vector_type(8))) float v8f;

#define DHEAD 64
#define NEGV  (-1e9f)
#define INVT  (0.125f)   // 1/TEMPERATURE, TEMPERATURE = 8
#define QSTR  68         // LDS row stride for 64-float rows: 16B-aligned, bank-conflict free
#define PSTR  20         // LDS row stride for 16-float rows: 16B-aligned, bank-conflict free

__device__ __forceinline__ v8f wmma_f32(v2f a, v2f b, v8f c) {
  // V_WMMA_F32_16X16X4_F32 : D = A(16x4) * B(4x16) + C(16x16)
  return __builtin_amdgcn_wmma_f32_16x16x4_f32(
      /*neg_a=*/false, a, /*neg_b=*/false, b,
      /*c_mod=*/(short)0, c, /*reuse_a=*/false, /*reuse_b=*/false);
}

// ---------------------------------------------------------------------------
// Kernel 1: scores = (Q/T) @ K^T, optional mask -> NEG. One wave per 16x16
// score tile; 8 waves per block cover a 16(q) x 128(k) region.
// grid = (ceil(Sk/128), ceil(Sq/16), B*H), block = 256
// ---------------------------------------------------------------------------
__global__ __launch_bounds__(256)
void scores_kernel(const float* __restrict__ Q, const float* __restrict__ K,
                   float* __restrict__ attn, const int* __restrict__ mask,
                   int Sq, int Sk, int H)
{
  __shared__ float qs[16 * QSTR];
  __shared__ float ks[8][16 * QSTR];

  const int tid  = threadIdx.x;
  const int w    = tid >> 5;
  const int lane = tid & 31;
  const int half = lane >> 4;
  const int l16  = lane & 15;

  const int bh = blockIdx.z;
  const int q0 = blockIdx.y * 16;
  const int k0 = blockIdx.x * 128 + w * 16;

  const float* Qb = Q + (size_t)bh * Sq * DHEAD;
  const float* Kb = K + (size_t)bh * Sk * DHEAD;
  float*       Ab = attn + (size_t)bh * Sq * Sk;

  // Stage Q tile (pre-scaled by 1/T) with b128 loads; 256 float4 / 256 threads.
  {
    const int r = tid >> 4, c4 = tid & 15;
    float4 v = {0.f, 0.f, 0.f, 0.f};
    if (q0 + r < Sq) {
      v = *(const float4*)&Qb[(size_t)(q0 + r) * DHEAD + c4 * 4];
      v.x *= INVT; v.y *= INVT; v.z *= INVT; v.w *= INVT;
    }
    *(float4*)&qs[r * QSTR + c4 * 4] = v;
  }
  // Stage this wave's K tile with b128 loads; 256 float4 / 32 lanes = 8 each.
#pragma unroll
  for (int t = 0; t < 8; ++t) {
    const int j = lane + t * 32;
    const int r = j >> 4, c4 = j & 15;
    float4 v = {0.f, 0.f, 0.f, 0.f};
    if (k0 + r < Sk) v = *(const float4*)&Kb[(size_t)(k0 + r) * DHEAD + c4 * 4];
    *(float4*)&ks[w][r * QSTR + c4 * 4] = v;
  }
  __syncthreads();

  v8f acc = {0.f, 0.f, 0.f, 0.f, 0.f, 0.f, 0.f, 0.f};
#pragma unroll
  for (int kk = 0; kk < DHEAD / 4; ++kk) {
    const int kb = kk * 4 + half * 2;
    v2f a, b;
    a.x = qs[l16 * QSTR + kb];         // A[m=l16][k]
    a.y = qs[l16 * QSTR + kb + 1];
    b.x = ks[w][l16 * QSTR + kb];      // B[k][n=l16] = K[n][k]
    b.y = ks[w][l16 * QSTR + kb + 1];
    acc = wmma_f32(a, b, acc);
  }

  const int b_idx = bh / H;
#pragma unroll
  for (int r = 0; r < 8; ++r) {
    const int m   = r + half * 8;
    const int row = q0 + m;
    const int col = k0 + l16;
    if (row < Sq && col < Sk) {
      float v = acc[r];
      if (mask) {
        if (mask[(size_t)b_idx * Sq * Sk + (size_t)row * Sk + col] == 0) v = NEGV;
      }
      Ab[(size_t)row * Sk + col] = v;
    }
  }
}

// ---------------------------------------------------------------------------
// Kernel 2: row softmax in place, single global read per element (values
// cached in registers: Sk <= 2048 -> <= 8 per thread).
// grid = B*H*Sq rows, block = 256
// ---------------------------------------------------------------------------
__global__ __launch_bounds__(256)
void softmax_kernel(float* __restrict__ attn, int Sk)
{
  __shared__ float red[256];
  const int tid = threadIdx.x;
  float* row = attn + (size_t)blockIdx.x * Sk;

  float vals[8];
  int   cnt = 0;
  float m = -3.4e38f;
  for (int i = tid; i < Sk; i += 256) {
    const float v = row[i];
    vals[cnt++] = v;
    m = fmaxf(m, v);
  }
  red[tid] = m; __syncthreads();
  for (int s = 128; s > 0; s >>= 1) {
    if (tid < s) red[tid] = fmaxf(red[tid], red[tid + s]);
    __syncthreads();
  }
  m = red[0]; __syncthreads();

  float sum = 0.0f;
#pragma unroll
  for (int j = 0; j < 8; ++j) {
    if (j < cnt) {
      vals[j] = __expf(vals[j] - m);
      sum += vals[j];
    }
  }
  red[tid] = sum; __syncthreads();
  for (int s = 128; s > 0; s >>= 1) {
    if (tid < s) red[tid] += red[tid + s];
    __syncthreads();
  }
  const float inv = 1.0f / red[0];

  cnt = 0;
  for (int i = tid; i < Sk; i += 256) row[i] = vals[cnt++] * inv;
}

// ---------------------------------------------------------------------------
// Kernel 3: out = attnA (+ attnB) @ V  (accumulating into out when doAcc).
// 4 waves/block, each owns a 16-row q-tile; V tile shared through LDS.
// grid = (ceil(Sq/64), B*H), block = 128
// ---------------------------------------------------------------------------
__global__ __launch_bounds__(128)
void pv_kernel(const float* __restrict__ attnA, const float* __restrict__ attnB,
               const float* __restrict__ V, float* __restrict__ out,
               int Sq, int Sk, int vRows, int outRows, int rowOff, int doAcc)
{
  __shared__ float vt[16 * QSTR];        // [k][d]
  __shared__ float pt[4][16 * PSTR];     // per-wave prob tile

  const int tid  = threadIdx.x;
  const int w    = tid >> 5;
  const int lane = tid & 31;
  const int half = lane >> 4;
  const int l16  = lane & 15;

  const int bh = blockIdx.y;
  const int q0 = (blockIdx.x * 4 + w) * 16;

  const float* Aa = attnA + (size_t)bh * Sq * Sk;
  const float* Bb = attnB ? (attnB + (size_t)bh * Sq * Sk) : nullptr;
  const float* Vb = V + (size_t)bh * vRows * DHEAD;
  float*       Ob = out + (size_t)bh * outRows * DHEAD;

  v8f acc[4];
#pragma unroll
  for (int nt = 0; nt < 4; ++nt) acc[nt] = (v8f){0.f,0.f,0.f,0.f,0.f,0.f,0.f,0.f};

  const int nChunk = (Sk + 15) / 16;
  for (int kc = 0; kc < nChunk; ++kc) {
    const int kbase = kc * 16;

    // Prefetch next V tile (speculative; emits global_prefetch_b8).
    if (kbase + 16 < Sk)
      __builtin_prefetch(&Vb[(size_t)(kbase + 16) * DHEAD + tid * 8], 0, 1);

    // Stage V tile with b128 loads; 256 float4 / 128 threads = 2 each.
#pragma unroll
    for (int t = 0; t < 2; ++t) {
      const int j = tid + t * 128;
      const int r = j >> 4, c4 = j & 15;
      float4 v = {0.f, 0.f, 0.f, 0.f};
      if (kbase + r < Sk) v = *(const float4*)&Vb[(size_t)(kbase + r) * DHEAD + c4 * 4];
      *(float4*)&vt[r * QSTR + c4 * 4] = v;
    }
    // Stage this wave's probability tile (scalar: attn rows are not 16B-aligned).
#pragma unroll
    for (int t = 0; t < 8; ++t) {
      const int i = lane + t * 32;
      const int r = i >> 4, c = i & 15;
      const int row = q0 + r, col = kbase + c;
      float p = 0.0f;
      if (row < Sq && col < Sk) {
        p = Aa[(size_t)row * Sk + col];
        if (Bb) p += Bb[(size_t)row * Sk + col];
      }
      pt[w][r * PSTR + c] = p;
    }
    __syncthreads();

#pragma unroll
    for (int st = 0; st < 4; ++st) {
      const int kb = st * 4 + half * 2;
      v2f a;
      a.x = pt[w][l16 * PSTR + kb];        // A[m=l16][k]
      a.y = pt[w][l16 * PSTR + kb + 1];
#pragma unroll
      for (int nt = 0; nt < 4; ++nt) {
        v2f b;
        b.x = vt[kb * QSTR + nt * 16 + l16];        // B[k][n] = V[k][n]
        b.y = vt[(kb + 1) * QSTR + nt * 16 + l16];
        acc[nt] = wmma_f32(a, b, acc[nt]);
      }
    }
    __syncthreads();
  }

#pragma unroll
  for (int nt = 0; nt < 4; ++nt) {
#pragma unroll
    for (int r = 0; r < 8; ++r) {
      const int m   = r + half * 8;
      const int row = q0 + m;
      if (row < Sq) {
        const size_t idx = (size_t)(row + rowOff) * DHEAD + nt * 16 + l16;
        const float v = acc[nt][r];
        Ob[idx] = doAcc ? (Ob[idx] + v) : v;
      }
    }
  }
}

// ---------------------------------------------------------------------------
extern "C" void kernel_launch(void* const* d_in, const int* in_sizes, int n_in,
                              void* d_out, int out_size, void* d_ws, size_t ws_size,
                              hipStream_t stream) {
  (void)in_sizes; (void)n_in; (void)out_size; (void)d_ws; (void)ws_size;

  const int B = 2, H = 8, S = 2048, Sm = 2047, BH = B * H;

  const float* poi_q  = (const float*)d_in[0];
  const float* poi_k  = (const float*)d_in[1];
  const float* poi_v  = (const float*)d_in[2];
  const float* time_q = (const float*)d_in[3];
  const float* time_k = (const float*)d_in[4];
  const float* time_v = (const float*)d_in[5];
  const float* dist_q = (const float*)d_in[6];
  const float* dist_k = (const float*)d_in[7];
  const float* dist_v = (const float*)d_in[8];
  const int*   mask   = (const int*)d_in[9];

  float* out = (float*)d_out;
  const size_t o_poi  = 0;
  const size_t o_time = o_poi  + (size_t)BH * S  * DHEAD;
  const size_t o_dist = o_time + (size_t)BH * Sm * DHEAD;
  const size_t o_a1   = o_dist + (size_t)BH * Sm * DHEAD;
  const size_t o_a2   = o_a1   + (size_t)BH * S  * S;
  const size_t o_a3   = o_a2   + (size_t)BH * Sm * Sm;

  float* poi_out  = out + o_poi;
  float* time_out = out + o_time;
  float* dist_out = out + o_dist;
  float* attn1    = out + o_a1;
  float* attn2    = out + o_a2;
  float* attn3    = out + o_a3;

  // --- scores ---
  {
    dim3 g1((S  + 127) / 128, (S  + 15) / 16, BH);
    dim3 g2((Sm + 127) / 128, (Sm + 15) / 16, BH);
    scores_kernel<<<g1, 256, 0, stream>>>(poi_q,  poi_k,  attn1, mask,    S,  S,  H);
    scores_kernel<<<g2, 256, 0, stream>>>(time_q, time_k, attn2, nullptr, Sm, Sm, H);
    scores_kernel<<<g2, 256, 0, stream>>>(dist_q, dist_k, attn3, nullptr, Sm, Sm, H);
  }

  // --- softmax (in place) ---
  softmax_kernel<<<BH * S,  256, 0, stream>>>(attn1, S);
  softmax_kernel<<<BH * Sm, 256, 0, stream>>>(attn2, Sm);
  softmax_kernel<<<BH * Sm, 256, 0, stream>>>(attn3, Sm);

  // --- PV ---
  {
    dim3 gp(((S  + 15) / 16 + 3) / 4, BH);   // 32 x 16
    dim3 gm(((Sm + 15) / 16 + 3) / 4, BH);   // 32 x 16
    pv_kernel<<<gp, 128, 0, stream>>>(attn1, nullptr, poi_v,  poi_out,  S,  S,  S,  S,  0, 0);
    pv_kernel<<<gm, 128, 0, stream>>>(attn2, nullptr, time_v, time_out, Sm, Sm, Sm, Sm, 0, 0);
    pv_kernel<<<gm, 128, 0, stream>>>(attn3, nullptr, dist_v, dist_out, Sm, Sm, Sm, Sm, 0, 0);
    // cross: poi_out[1:] += (attn2 + attn3) @ poi_v[:-1]  (ordered after base PV)
    pv_kernel<<<gm, 128, 0, stream>>>(attn2, attn3,  poi_v,  poi_out,  Sm, Sm, S,  S,  1, 1);
  }
}